// Eagle3LlamaAttention_3805341024311
// MI455X (gfx1250) — compile-verified
//
#include <hip/hip_runtime.h>

typedef __attribute__((ext_vector_type(2))) float v2f;
typedef __attribute__((ext_vector_type(8))) float v8f;
typedef int vi4 __attribute__((vector_size(16)));

#define B_    2
#define T_    2048
#define H_    2048
#define NH_   32
#define NKV_  8
#define DH_   64
#define STEP_ 2
#define KDIM_ 4096          // 2*H
#define NEGV  (-1000000000.0f)
#define THETA 10000.0f

#if __has_builtin(__builtin_amdgcn_global_load_async_to_lds_b128) && \
    __has_builtin(__builtin_amdgcn_global_load_async_to_lds_b32)  && \
    __has_builtin(__builtin_amdgcn_s_wait_asynccnt)
#define USE_ASYNC 1
#else
#define USE_ASYNC 0
#endif

// Copy 4B / 16B from global to LDS. Async path uses the CDNA5 async-DMA
// (ASYNCcnt-tracked) engine; per-lane LDS address is independent of the
// per-lane global address, so transposed LDS layouts are free.
__device__ __forceinline__ void cp_b32(float* lds, const float* g) {
#if USE_ASYNC
  __builtin_amdgcn_global_load_async_to_lds_b32(
      (__attribute__((address_space(1))) int*)(void*)g,
      (__attribute__((address_space(3))) int*)(void*)lds, 0, 0);
#else
  *lds = *g;
#endif
}
__device__ __forceinline__ void cp_b128(float* lds, const float* g) {
#if USE_ASYNC
  __builtin_amdgcn_global_load_async_to_lds_b128(
      (__attribute__((address_space(1))) vi4*)(void*)g,
      (__attribute__((address_space(3))) vi4*)(void*)lds, 0, 0);
#else
  *(float4*)lds = *(const float4*)g;
#endif
}
__device__ __forceinline__ void async_wait() {
#if USE_ASYNC
  __builtin_amdgcn_s_wait_asynccnt(0);
#endif
}

__device__ __forceinline__ v8f wmma_f32(v2f a, v2f b, v8f c) {
  return __builtin_amdgcn_wmma_f32_16x16x4_f32(
      false, a, false, b, (short)0, c, false, false);
}

__device__ __forceinline__ void scale_v8(v8f& v, float a) {
#pragma unroll
  for (int i = 0; i < 8; ++i) v[i] *= a;
}

#define XS_STR 68   // 16x64 tile, padded: bank = 4*row + k  -> conflict-free
#define WS_STR 68   // transposed weight tile, same property
#define VS_STR 20   // VsT[dh][key], 16+4 pad
#define PS_STR 20   // PsT[q][key]

// ---------------------------------------------------------------------------
// Kernel 1: fused QKV GEMM (+RoPE epilogue), LDS-tiled, async-staged.
// Block 128 (4 waves): 64 M-rows x 64 N-cols; W strip shared by all 4 waves.
// grid (48, 64).
// ---------------------------------------------------------------------------
__global__ __launch_bounds__(128)
void qkv_rope_kernel(const float* __restrict__ X,
                     const float* __restrict__ Wq,
                     const float* __restrict__ Wk,
                     const float* __restrict__ Wv,
                     float* __restrict__ qb,
                     float* __restrict__ kb,
                     float* __restrict__ vb) {
  __shared__ float Xs[64][XS_STR];     // [wave-tile*16 + r][k]
  __shared__ float WsT[64][WS_STR];    // [col][k]  (transposed)

  const int tid  = threadIdx.x;
  const int lane = tid & 31;
  const int wave = tid >> 5;
  const int hl   = lane >> 4;
  const int mrow = lane & 15;
  const int mt   = blockIdx.y * 4 + wave;
  const int n0   = blockIdx.x * 64;
  const int mbase = blockIdx.y * 64;

  const float* W; int ldw, cbase, seg;
  if (n0 < 2048)      { W = Wq; ldw = 2048; cbase = n0;        seg = 0; }
  else if (n0 < 2560) { W = Wk; ldw = 512;  cbase = n0 - 2048; seg = 1; }
  else                { W = Wv; ldw = 512;  cbase = n0 - 2560; seg = 2; }

  v8f acc0 = {}, acc1 = {}, acc2 = {}, acc3 = {};

  for (int k0 = 0; k0 < KDIM_; k0 += 64) {
    // Stage X tile: 64 rows x 64 k, natural layout, b128 (coalesced).
    for (int e4 = tid; e4 < 1024; e4 += 128) {
      const int row = e4 >> 4;
      const int c4  = (e4 & 15) * 4;
      cp_b128(&Xs[row][c4], X + (size_t)(mbase + row) * KDIM_ + k0 + c4);
    }
    // Stage W^T tile: global reads coalesced along columns, LDS transposed.
    for (int e = tid; e < 4096; e += 128) {
      const int k = e >> 6, c = e & 63;
      cp_b32(&WsT[c][k], W + (size_t)(k0 + k) * ldw + cbase + c);
    }
    async_wait();
    __syncthreads();

#pragma unroll
    for (int kk = 0; kk < 16; ++kk) {
      const int k4 = kk * 4 + 2 * hl;
      v2f a  = *(const v2f*)&Xs[wave * 16 + mrow][k4];
      v2f b0 = *(const v2f*)&WsT[ 0 + mrow][k4];
      v2f b1 = *(const v2f*)&WsT[16 + mrow][k4];
      v2f b2 = *(const v2f*)&WsT[32 + mrow][k4];
      v2f b3 = *(const v2f*)&WsT[48 + mrow][k4];
      acc0 = wmma_f32(a, b0, acc0);
      acc1 = wmma_f32(a, b1, acc1);
      acc2 = wmma_f32(a, b2, acc2);
      acc3 = wmma_f32(a, b3, acc3);
    }
    __syncthreads();
  }

  if (seg == 2) {                       // V: plain transposed store
    const int head = cbase / 64;
#pragma unroll
    for (int r = 0; r < 8; ++r) {
      const int gm = mt * 16 + r + 8 * hl;
      const int b  = gm / T_, t = gm % T_;
      float* o = vb + (((size_t)(b * NKV_ + head) * T_ + t) * DH_);
      o[ 0 + mrow] = acc0[r];
      o[16 + mrow] = acc1[r];
      o[32 + mrow] = acc2[r];
      o[48 + mrow] = acc3[r];
    }
  } else {                              // Q or K: fuse RoPE
    const int head    = cbase / 64;
    const int nheads  = (seg == 0) ? NH_ : NKV_;
    float* outp       = (seg == 0) ? qb : kb;
    const float d0 = (float)(mrow);
    const float d1 = (float)(16 + mrow);
    const float if0 = __powf(THETA, -d0 / 32.0f);
    const float if1 = __powf(THETA, -d1 / 32.0f);
#pragma unroll
    for (int r = 0; r < 8; ++r) {
      const int gm = mt * 16 + r + 8 * hl;
      const int b  = gm / T_, t = gm % T_;
      const float pos = (float)(t + STEP_);
      float c0, s0, c1, s1;
      __sincosf(pos * if0, &s0, &c0);
      __sincosf(pos * if1, &s1, &c1);
      const float xl0 = acc0[r], xh0 = acc2[r];
      const float xl1 = acc1[r], xh1 = acc3[r];
      float* o = outp + (((size_t)(b * nheads + head) * T_ + t) * DH_);
      o[ 0 + mrow] = xl0 * c0 - xh0 * s0;
      o[32 + mrow] = xh0 * c0 + xl0 * s0;
      o[16 + mrow] = xl1 * c1 - xh1 * s1;
      o[48 + mrow] = xh1 * c1 + xl1 * s1;
    }
  }
}

// ---------------------------------------------------------------------------
// Kernel 2: flash attention + 2 diagonal columns. One wave per
// (b, h, 16-query tile). S^T = K*Q^T, O^T = V^T*P^T, all WMMA f32.
// grid (T/16, NH, B), block 32.
// ---------------------------------------------------------------------------
__global__ __launch_bounds__(32)
void attn_kernel(const float* __restrict__ qb,
                 const float* __restrict__ kbuf,
                 const float* __restrict__ vbuf,
                 const float* __restrict__ ck,
                 const float* __restrict__ cv,
                 float* __restrict__ O) {
  __shared__ float Qs[16][XS_STR];
  __shared__ float Ks[16][XS_STR];
  __shared__ float VsT[64][VS_STR];    // [dh][key] (transposed)
  __shared__ float PsT[16][PS_STR];    // [q][key]  (transposed)

  const int lane = threadIdx.x;
  const int hl   = lane >> 4;
  const int mrow = lane & 15;
  const int qt = blockIdx.x;
  const int h  = blockIdx.y;
  const int b  = blockIdx.z;
  const int hk = h / (NH_ / NKV_);
  const float scale = 0.125f;

  const float* ck0 = ck + (((size_t)b * NH_ + h) * T_) * (size_t)DH_;
  const float* ck1 = ck + ((((size_t)B_ + b) * NH_ + h) * T_) * (size_t)DH_;
  const float* cv0 = cv + (((size_t)b * NH_ + h) * T_) * (size_t)DH_;
  const float* cv1 = cv + ((((size_t)B_ + b) * NH_ + h) * T_) * (size_t)DH_;

  // Stage Q tile (16x64) via async b128
  const float* Qg = qb + (((size_t)b * NH_ + h) * T_ + qt * 16) * (size_t)DH_;
  for (int e4 = lane; e4 < 256; e4 += 32) {
    const int r = e4 >> 4, c4 = (e4 & 15) * 4;
    cp_b128(&Qs[r][c4], Qg + r * 64 + c4);
  }

  float m = -3.0e38f, l = 0.0f;
  v8f o0 = {}, o1 = {}, o2 = {}, o3 = {};

  const int nkb = qt + 1;
  for (int kbi = 0; kbi < nkb; ++kbi) {
    const float* Kg = ck0 + (size_t)kbi * 16 * DH_;
    const float* Vg = cv0 + (size_t)kbi * 16 * DH_;
    // K: natural layout, b128; V: transposed, b32 (free with async DMA)
    for (int e4 = lane; e4 < 256; e4 += 32) {
      const int r = e4 >> 4, c4 = (e4 & 15) * 4;
      cp_b128(&Ks[r][c4], Kg + r * 64 + c4);
    }
    for (int e = lane; e < 1024; e += 32) {
      const int key = e >> 6, dh = e & 63;
      cp_b32(&VsT[dh][key], Vg + e);
    }
    async_wait();

    // S^T(16k x 16q) = K(16x64) * Q^T(64x16)
    v8f s_acc = {};
#pragma unroll
    for (int kk = 0; kk < 16; ++kk) {
      const int k4 = kk * 4 + 2 * hl;
      v2f a  = *(const v2f*)&Ks[mrow][k4];
      v2f bq = *(const v2f*)&Qs[mrow][k4];
      s_acc = wmma_f32(a, bq, s_acc);
    }

    float sv[8];
    const int t = qt * 16 + mrow;
#pragma unroll
    for (int r = 0; r < 8; ++r) {
      float s = s_acc[r] * scale;
      const int key = kbi * 16 + r + 8 * hl;
      if (key > t) s += NEGV;
      sv[r] = s;
    }
    float bmax = sv[0];
#pragma unroll
    for (int r = 1; r < 8; ++r) bmax = fmaxf(bmax, sv[r]);
    bmax = fmaxf(bmax, __shfl_xor(bmax, 16, 32));
    const float m_new = fmaxf(m, bmax);
    const float alpha = __expf(m - m_new);
    float psum = 0.0f;
#pragma unroll
    for (int r = 0; r < 8; ++r) { sv[r] = __expf(sv[r] - m_new); psum += sv[r]; }
    psum += __shfl_xor(psum, 16, 32);
    l = l * alpha + psum;
    m = m_new;
    scale_v8(o0, alpha); scale_v8(o1, alpha);
    scale_v8(o2, alpha); scale_v8(o3, alpha);

    // relayout P (transposed): PsT[q][key]
#pragma unroll
    for (int r = 0; r < 8; ++r) PsT[mrow][r + 8 * hl] = sv[r];

    // O^T(64 x 16q) += V^T(64 x 16k) * P^T(16k x 16q)
#pragma unroll
    for (int dt = 0; dt < 4; ++dt) {
      v8f* od = (dt == 0) ? &o0 : (dt == 1) ? &o1 : (dt == 2) ? &o2 : &o3;
#pragma unroll
      for (int ks = 0; ks < 4; ++ks) {
        const int kidx = ks * 4 + 2 * hl;
        v2f a  = *(const v2f*)&VsT[dt * 16 + mrow][kidx];
        v2f bp = *(const v2f*)&PsT[mrow][kidx];
        *od = wmma_f32(a, bp, *od);
      }
    }
  }

  // ---- 2 extra diagonal columns ----
  const int t = qt * 16 + mrow;
  const float* qrow = Qs[mrow];
  const float* ke0  = ck1 + (size_t)t * DH_;
  const float* ke1  = kbuf + (((size_t)b * NKV_ + hk) * T_ + t) * (size_t)DH_;
  float e0 = 0.0f, e1 = 0.0f;
  for (int d = hl * 32; d < hl * 32 + 32; ++d) {
    e0 += qrow[d] * ke0[d];
    e1 += qrow[d] * ke1[d];
  }
  e0 += __shfl_xor(e0, 16, 32);
  e1 += __shfl_xor(e1, 16, 32);
  e0 *= scale; e1 *= scale;

  const float m_f   = fmaxf(m, fmaxf(e0, e1));
  const float alpha = __expf(m - m_f);
  const float p0 = __expf(e0 - m_f);
  const float p1 = __expf(e1 - m_f);
  l = l * alpha + p0 + p1;

  const float* ve0 = cv1 + (size_t)t * DH_;
  const float* ve1 = vbuf + (((size_t)b * NKV_ + hk) * T_ + t) * (size_t)DH_;
  const float rinv = 1.0f / l;
  float* Or = O + ((size_t)(b * T_ + t)) * (NH_ * DH_) + h * DH_;
#pragma unroll
  for (int dt = 0; dt < 4; ++dt) {
    v8f* od = (dt == 0) ? &o0 : (dt == 1) ? &o1 : (dt == 2) ? &o2 : &o3;
#pragma unroll
    for (int r = 0; r < 8; ++r) {
      const int dh = dt * 16 + r + 8 * hl;
      const float val = (*od)[r] * alpha + p0 * ve0[dh] + p1 * ve1[dh];
      Or[dh] = val * rinv;
    }
  }
}

// ---------------------------------------------------------------------------
// Kernel 3: output projection, LDS-tiled like kernel 1. grid (32, 64).
// ---------------------------------------------------------------------------
__global__ __launch_bounds__(128)
void out_gemm_kernel(const float* __restrict__ X,
                     const float* __restrict__ Wo,
                     float* __restrict__ Y) {
  __shared__ float Xs[64][XS_STR];
  __shared__ float WsT[64][WS_STR];

  const int tid  = threadIdx.x;
  const int lane = tid & 31;
  const int wave = tid >> 5;
  const int hl   = lane >> 4;
  const int mrow = lane & 15;
  const int mt   = blockIdx.y * 4 + wave;
  const int n0   = blockIdx.x * 64;
  const int mbase = blockIdx.y * 64;

  v8f acc0 = {}, acc1 = {}, acc2 = {}, acc3 = {};

  for (int k0 = 0; k0 < H_; k0 += 64) {
    for (int e4 = tid; e4 < 1024; e4 += 128) {
      const int row = e4 >> 4;
      const int c4  = (e4 & 15) * 4;
      cp_b128(&Xs[row][c4], X + (size_t)(mbase + row) * H_ + k0 + c4);
    }
    for (int e = tid; e < 4096; e += 128) {
      const int k = e >> 6, c = e & 63;
      cp_b32(&WsT[c][k], Wo + (size_t)(k0 + k) * H_ + n0 + c);
    }
    async_wait();
    __syncthreads();

#pragma unroll
    for (int kk = 0; kk < 16; ++kk) {
      const int k4 = kk * 4 + 2 * hl;
      v2f a  = *(const v2f*)&Xs[wave * 16 + mrow][k4];
      v2f b0 = *(const v2f*)&WsT[ 0 + mrow][k4];
      v2f b1 = *(const v2f*)&WsT[16 + mrow][k4];
      v2f b2 = *(const v2f*)&WsT[32 + mrow][k4];
      v2f b3 = *(const v2f*)&WsT[48 + mrow][k4];
      acc0 = wmma_f32(a, b0, acc0);
      acc1 = wmma_f32(a, b1, acc1);
      acc2 = wmma_f32(a, b2, acc2);
      acc3 = wmma_f32(a, b3, acc3);
    }
    __syncthreads();
  }

#pragma unroll
  for (int r = 0; r < 8; ++r) {
    const int gm = mt * 16 + r + 8 * hl;
    float* o = Y + (size_t)gm * H_ + n0;
    o[ 0 + mrow] = acc0[r];
    o[16 + mrow] = acc1[r];
    o[32 + mrow] = acc2[r];
    o[48 + mrow] = acc3[r];
  }
}

// ---------------------------------------------------------------------------
extern "C" void kernel_launch(void* const* d_in, const int* in_sizes, int n_in,
                              void* d_out, int out_size, void* d_ws, size_t ws_size,
                              hipStream_t stream) {
  (void)in_sizes; (void)n_in; (void)out_size; (void)ws_size;
  const float* cs  = (const float*)d_in[0];
  const float* ckp = (const float*)d_in[3];
  const float* cvp = (const float*)d_in[4];
  const float* Wq  = (const float*)d_in[5];
  const float* Wk  = (const float*)d_in[6];
  const float* Wv  = (const float*)d_in[7];
  const float* Wo  = (const float*)d_in[8];
  float* out = (float*)d_out;

  float* qb = (float*)d_ws;
  float* kb = qb + (size_t)B_ * NH_ * T_ * DH_;
  float* vb = kb + (size_t)B_ * NKV_ * T_ * DH_;
  float* ao = vb + (size_t)B_ * NKV_ * T_ * DH_;

  qkv_rope_kernel<<<dim3(48, 64), 128, 0, stream>>>(cs, Wq, Wk, Wv, qb, kb, vb);
  attn_kernel<<<dim3(T_ / 16, NH_, B_), 32, 0, stream>>>(qb, kb, vb, ckp, cvp, ao);
  out_gemm_kernel<<<dim3(H_ / 64, (B_ * T_) / 64), 128, 0, stream>>>(ao, Wo, out);
}